// MyPointTransformerConv_1219770712820
// MI455X (gfx1250) — compile-verified
//
#include <hip/hip_runtime.h>
#include <hip/hip_bf16.h>

#define D 128

typedef __attribute__((ext_vector_type(2))) float v2f;
typedef __attribute__((ext_vector_type(8))) float v8f;

// D(16x16,f32) = A(16x4,f32) x B(4x16,f32) + C
// A layout (ISA 7.12.2, 32-bit A 16x4): lanes 0-15: M=lane, VGPR0=K0,VGPR1=K1;
//                                       lanes 16-31: M=lane-16, VGPR0=K2,VGPR1=K3.
// B layout (symmetric): lanes 0-15: N=lane, K0/K1; lanes 16-31: N=lane-16, K2/K3.
// C/D: VGPR i -> M = i + (lane>=16 ? 8 : 0), N = lane&15.
__device__ __forceinline__ v8f wmma_f32_4(v2f a, v2f b, v8f c) {
  return __builtin_amdgcn_wmma_f32_16x16x4_f32(false, a, false, b, (short)0, c,
                                               false, false);
}

// ---------------------------------------------------------------------------
// Kernel 1: three node GEMMs  (x @ {W, Wsrc, Wdst}) -> {x_proj, a_src, a_dst}
// grid = (N/16, 3), block = 128 (4 waves); wave w computes col tiles 2w, 2w+1.
// ---------------------------------------------------------------------------
__global__ __launch_bounds__(128)
void node_gemm3_kernel(const float* __restrict__ x,
                       const float* __restrict__ W0,
                       const float* __restrict__ W1,
                       const float* __restrict__ W2,
                       float* __restrict__ o0, float* __restrict__ o1,
                       float* __restrict__ o2) {
  const float* Wm;
  float* out;
  if (blockIdx.y == 0)      { Wm = W0; out = o0; }
  else if (blockIdx.y == 1) { Wm = W1; out = o1; }
  else                      { Wm = W2; out = o2; }

  const int row0  = blockIdx.x * 16;
  const int lane  = threadIdx.x & 31;
  const int wave  = threadIdx.x >> 5;
  const int m     = lane & 15;
  const int khalf = (lane >> 4) << 1;   // 0 (lanes 0-15) or 2 (lanes 16-31)
  const int mbase = (lane >> 4) << 3;   // 0 or 8 for C/D rows

  const float* arow = x + (size_t)(row0 + m) * D;

  for (int t = 0; t < 2; ++t) {
    const int col0 = (wave * 2 + t) * 16;
    v8f c = {};
    for (int k0 = 0; k0 < D; k0 += 4) {
      const int ka = k0 + khalf;
      v2f a = *(const v2f*)(arow + ka);              // 8B-aligned global_load_b64
      v2f b;
      b.x = Wm[(size_t)ka * D + col0 + m];
      b.y = Wm[(size_t)(ka + 1) * D + col0 + m];
      c = wmma_f32_4(a, b, c);
    }
#pragma unroll
    for (int i = 0; i < 8; ++i)
      out[(size_t)(row0 + mbase + i) * D + col0 + m] = c[i];
  }
}

// ---------------------------------------------------------------------------
// Kernel 2: per-edge attention MLP + segment max.
// 16 edges per block; alpha0 and hidden tiles staged in LDS (row stride 132
// floats -> conflict-free A-frag ds_load_b64 on 64 banks).
// ---------------------------------------------------------------------------
__global__ __launch_bounds__(128)
void edge_alpha_kernel(const float* __restrict__ a_src,
                       const float* __restrict__ a_dst,
                       const float* __restrict__ pos,
                       const int*   __restrict__ ei,
                       const float* __restrict__ Wpos,
                       const float* __restrict__ bpos,
                       const float* __restrict__ Wa1,
                       const float* __restrict__ ba1,
                       const float* __restrict__ Wa2,
                       const float* __restrict__ ba2,
                       float* __restrict__ alpha_out,   // E x 128
                       int*   __restrict__ seg_max) {   // N x 128, int-bit f32 max
  __shared__ float sA[16 * 132];
  __shared__ float sH[16 * 132];
  __shared__ int   srow[16], scol[16];
  __shared__ float sdp[16][3];

  const int tid = threadIdx.x;
  const int e0  = blockIdx.x * 16;

  if (tid < 16) {
    srow[tid] = ei[(size_t)(e0 + tid) * 2 + 0];
    scol[tid] = ei[(size_t)(e0 + tid) * 2 + 1];
  }
  __syncthreads();
  if (tid < 48) {
    const int le = tid / 3, k = tid % 3;
    sdp[le][k] = pos[(size_t)srow[le] * 3 + k] - pos[(size_t)scol[le] * 3 + k];
  }
  const float wp0 = Wpos[tid], wp1 = Wpos[128 + tid], wp2 = Wpos[256 + tid];
  const float bp  = bpos[tid];
  __syncthreads();

  // alpha0[le][tid] = a_src[row] - a_dst[col] + delta   (coalesced gathers, L2-resident)
  for (int le = 0; le < 16; ++le) {
    const int r = srow[le], cc = scol[le];
    const float d = sdp[le][0] * wp0 + sdp[le][1] * wp1 + sdp[le][2] * wp2 + bp;
    sA[le * 132 + tid] =
        a_src[(size_t)r * D + tid] - a_dst[(size_t)cc * D + tid] + d;
  }
  __syncthreads();

  const int lane  = tid & 31;
  const int wave  = tid >> 5;
  const int m     = lane & 15;
  const int khalf = (lane >> 4) << 1;
  const int mbase = (lane >> 4) << 3;

  // GEMM1: h = relu(alpha0 @ Wa1 + ba1)
  for (int t = 0; t < 2; ++t) {
    const int col0 = (wave * 2 + t) * 16;
    v8f c = {};
    for (int k0 = 0; k0 < D; k0 += 4) {
      const int ka = k0 + khalf;
      v2f a = *(const v2f*)(&sA[m * 132 + ka]);      // ds_load_b64
      v2f b;
      b.x = Wa1[(size_t)ka * D + col0 + m];
      b.y = Wa1[(size_t)(ka + 1) * D + col0 + m];
      c = wmma_f32_4(a, b, c);
    }
    const float bb = ba1[col0 + m];
#pragma unroll
    for (int i = 0; i < 8; ++i) {
      const float v = c[i] + bb;
      sH[(mbase + i) * 132 + col0 + m] = v > 0.f ? v : 0.f;
    }
  }
  __syncthreads();

  // GEMM2: alpha = h @ Wa2 + ba2 ; store + segment max (positive-only int max)
  for (int t = 0; t < 2; ++t) {
    const int col0 = (wave * 2 + t) * 16;
    v8f c = {};
    for (int k0 = 0; k0 < D; k0 += 4) {
      const int ka = k0 + khalf;
      v2f a = *(const v2f*)(&sH[m * 132 + ka]);
      v2f b;
      b.x = Wa2[(size_t)ka * D + col0 + m];
      b.y = Wa2[(size_t)(ka + 1) * D + col0 + m];
      c = wmma_f32_4(a, b, c);
    }
    const float bb = ba2[col0 + m];
#pragma unroll
    for (int i = 0; i < 8; ++i) {
      const int   em = mbase + i;
      const float v  = c[i] + bb;
      alpha_out[(size_t)(e0 + em) * D + col0 + m] = v;
      // seg_max init 0.0; final = max(0, max alpha). Positive floats are
      // monotone under signed-int compare, so int atomicMax is exact.
      if (v > 0.f)
        atomicMax(&seg_max[(size_t)srow[em] * D + col0 + m],
                  __float_as_int(v));
    }
  }
}

// ---------------------------------------------------------------------------
// Kernel 3: exp(alpha - seg_max[row]) in place + segment sum (f32 atomics).
// ---------------------------------------------------------------------------
__global__ __launch_bounds__(128)
void edge_exp_kernel(const int* __restrict__ ei,
                     const float* __restrict__ seg_max,
                     float* __restrict__ alpha,
                     float* __restrict__ seg_sum) {
  const int e = blockIdx.x, t = threadIdx.x;
  const int r = ei[(size_t)e * 2];
  const size_t ia = (size_t)e * D + t;
  const float v = expf(alpha[ia] - seg_max[(size_t)r * D + t]);
  alpha[ia] = v;
  atomicAdd(&seg_sum[(size_t)r * D + t], v);
}

// ---------------------------------------------------------------------------
// Kernel 4: att = exp/(sum+eps); msg = att*(x_proj[col]+delta); scatter-add.
// delta is recomputed (3 FMAs) instead of storing 256 MB.
// ---------------------------------------------------------------------------
__global__ __launch_bounds__(128)
void edge_scatter_kernel(const int* __restrict__ ei,
                         const float* __restrict__ pos,
                         const float* __restrict__ Wpos,
                         const float* __restrict__ bpos,
                         const float* __restrict__ x_proj,
                         const float* __restrict__ alpha,
                         const float* __restrict__ seg_sum,
                         float* __restrict__ out) {
  const int e = blockIdx.x, t = threadIdx.x;
  const int r = ei[(size_t)e * 2], cc = ei[(size_t)e * 2 + 1];
  const float dpx = pos[(size_t)r * 3 + 0] - pos[(size_t)cc * 3 + 0];
  const float dpy = pos[(size_t)r * 3 + 1] - pos[(size_t)cc * 3 + 1];
  const float dpz = pos[(size_t)r * 3 + 2] - pos[(size_t)cc * 3 + 2];
  const float delta =
      dpx * Wpos[t] + dpy * Wpos[128 + t] + dpz * Wpos[256 + t] + bpos[t];
  const float att =
      alpha[(size_t)e * D + t] / (seg_sum[(size_t)r * D + t] + 1e-12f);
  const float msg = att * (x_proj[(size_t)cc * D + t] + delta);
  atomicAdd(&out[(size_t)r * D + t], msg);
}

// ---------------------------------------------------------------------------
extern "C" void kernel_launch(void* const* d_in, const int* in_sizes, int n_in,
                              void* d_out, int out_size, void* d_ws,
                              size_t ws_size, hipStream_t stream) {
  const float* x    = (const float*)d_in[0];
  const float* pos  = (const float*)d_in[1];
  const int*   ei   = (const int*)d_in[2];
  const float* W    = (const float*)d_in[3];
  const float* Wsrc = (const float*)d_in[4];
  const float* Wdst = (const float*)d_in[5];
  const float* Wpos = (const float*)d_in[6];
  const float* bpos = (const float*)d_in[7];
  const float* Wa1  = (const float*)d_in[8];
  const float* ba1  = (const float*)d_in[9];
  const float* Wa2  = (const float*)d_in[10];
  const float* ba2  = (const float*)d_in[11];

  const int N = in_sizes[0] / D;   // 50000 (multiple of 16)
  const int E = in_sizes[2] / 2;   // 500000 (multiple of 16)

  // Workspace layout (floats): needs (5N + E)*128*4 bytes (~384 MB).
  float* ws      = (float*)d_ws;
  float* x_proj  = ws;
  float* a_src   = x_proj + (size_t)N * D;
  float* a_dst   = a_src + (size_t)N * D;
  float* seg_max = a_dst + (size_t)N * D;
  float* seg_sum = seg_max + (size_t)N * D;
  float* alpha_e = seg_sum + (size_t)N * D;
  float* outf    = (float*)d_out;

  // seg_max and seg_sum are contiguous; zero both (and the poisoned output).
  hipMemsetAsync(seg_max, 0, (size_t)2 * N * D * sizeof(float), stream);
  hipMemsetAsync(outf, 0, (size_t)out_size * sizeof(float), stream);

  dim3 g1(N / 16, 3);
  node_gemm3_kernel<<<g1, 128, 0, stream>>>(x, W, Wsrc, Wdst, x_proj, a_src,
                                            a_dst);
  edge_alpha_kernel<<<E / 16, 128, 0, stream>>>(a_src, a_dst, pos, ei, Wpos,
                                                bpos, Wa1, ba1, Wa2, ba2,
                                                alpha_e, (int*)seg_max);
  edge_exp_kernel<<<E, 128, 0, stream>>>(ei, seg_max, alpha_e, seg_sum);
  edge_scatter_kernel<<<E, 128, 0, stream>>>(ei, pos, Wpos, bpos, x_proj,
                                             alpha_e, seg_sum, outf);
}